// TwoLayerGCN_10608569221485
// MI455X (gfx1250) — compile-verified
//
#include <hip/hip_runtime.h>
#include <hip/hip_bf16.h>

typedef __attribute__((ext_vector_type(2))) float v2f;
typedef __attribute__((ext_vector_type(4))) float v4f;
typedef __attribute__((ext_vector_type(8))) float v8f;

#define GB 8
#define GN 2048
#define GD 64
#define GC 16

__device__ __forceinline__ unsigned hashu(unsigned x) {
    x ^= x >> 17; x *= 0xed5ad4bbu;
    x ^= x >> 11; x *= 0xac4c1b51u;
    x ^= x >> 15; x *= 0x31848babu;
    x ^= x >> 14;
    return x;
}

// Issue one per-lane 16B global -> LDS async copy (ASYNCcnt-tracked, no VGPR staging)
__device__ __forceinline__ void async_g2l_b128(unsigned lds_addr, const void* gaddr) {
    asm volatile("global_load_async_to_lds_b128 %0, %1, off"
                 :: "v"(lds_addr), "v"(gaddr) : "memory");
}
__device__ __forceinline__ void wait_async0() {
    asm volatile("s_wait_asynccnt 0x0" ::: "memory");
}

// ---------------------------------------------------------------------------
// K1: dis[b*N+i] = rsqrt(1 + sum_j A[b,i,j])   (one wave per row)
// ---------------------------------------------------------------------------
__global__ __launch_bounds__(256) void k_degree(const float* __restrict__ A,
                                                float* __restrict__ dis) {
    const int lane = threadIdx.x & 31;
    const int row  = blockIdx.x * 8 + (threadIdx.x >> 5);   // [0, B*N)
    const float* p = A + (size_t)row * GN;
    float s = 0.0f;
    #pragma unroll 4
    for (int j = lane * 4; j < GN; j += 32 * 4) {
        v4f v = *(const v4f*)(p + j);
        s += v.x + v.y + v.z + v.w;
    }
    #pragma unroll
    for (int off = 16; off > 0; off >>= 1) s += __shfl_down(s, off, 32);
    if (lane == 0) dis[row] = rsqrtf(1.0f + s);
}

// ---------------------------------------------------------------------------
// K2: Z1p = rowscale(dis) * (X @ W1)   M=16384 K=64 N=64, WMMA f32 16x16x4
// ---------------------------------------------------------------------------
__global__ __launch_bounds__(256) void k_xw1(const float* __restrict__ X,
                                             const float* __restrict__ W1,
                                             const float* __restrict__ dis,
                                             float* __restrict__ Z1p) {
    const int lane = threadIdx.x & 31;
    const int wv   = threadIdx.x >> 5;
    const int row0 = (blockIdx.x * 8 + wv) * 16;
    const int m  = lane & 15;
    const int kh = (lane >> 4) * 2;
    const int n  = lane & 15;
    const float* xrow = X + (size_t)(row0 + m) * GD + kh;
    v8f acc[4] = {};
    for (int k = 0; k < GD; k += 4) {
        v2f a = *(const v2f*)(xrow + k);
        #pragma unroll
        for (int t = 0; t < 4; ++t) {
            v2f bf;
            bf.x = W1[(k + kh)     * GD + t * 16 + n];
            bf.y = W1[(k + kh + 1) * GD + t * 16 + n];
            acc[t] = __builtin_amdgcn_wmma_f32_16x16x4_f32(
                false, a, false, bf, (short)0, acc[t], false, false);
        }
    }
    #pragma unroll
    for (int t = 0; t < 4; ++t)
        #pragma unroll
        for (int r = 0; r < 8; ++r) {
            int row = row0 + r + 8 * (lane >> 4);
            Z1p[(size_t)row * GD + t * 16 + n] = acc[t][r] * dis[row];
        }
}

// ---------------------------------------------------------------------------
// K3: Hd = dropout(relu(dis*(A@Z1p + Z1p) + b1))
//     per batch: 2048x2048 @ 2048x64, double-buffered async LDS slabs
// ---------------------------------------------------------------------------
#define KB3 128
__global__ __launch_bounds__(256) void k_layer1(const float* __restrict__ A,
                                                const float* __restrict__ Zp,
                                                const float* __restrict__ dis,
                                                const float* __restrict__ b1,
                                                float* __restrict__ Hd) {
    __shared__ float lz[2][KB3][80];              // pad 64->80: half-waves hit disjoint banks
    const int b     = blockIdx.x >> 4;
    const int mtile = blockIdx.x & 15;
    const int lane  = threadIdx.x & 31;
    const int wv    = threadIdx.x >> 5;
    const int row0  = mtile * 128 + wv * 16;
    const int m  = lane & 15;
    const int kh = (lane >> 4) * 2;
    const int n  = lane & 15;
    const float* arow  = A  + ((size_t)b * GN + row0 + m) * GN + kh;
    const float* zbase = Zp + (size_t)b * GN * GD;
    const unsigned lbase = (unsigned)(uintptr_t)(void*)&lz[0][0][0];

    auto fill = [&](int buf, int k0) {
        const float* src = zbase + (size_t)k0 * GD;
        const unsigned lb = lbase + (unsigned)buf * (KB3 * 80 * 4);
        #pragma unroll
        for (int it = 0; it < 8; ++it) {          // 2048 x b128 per slab, 8 per thread
            int idx = (int)threadIdx.x + it * 256;
            int r = idx >> 4, c = idx & 15;
            async_g2l_b128(lb + (unsigned)(r * 320 + c * 16), src + r * GD + c * 4);
        }
    };

    v8f acc[4] = {};
    fill(0, 0);
    wait_async0();
    __syncthreads();
    for (int k0 = 0; k0 < GN; k0 += KB3) {
        const int buf = (k0 / KB3) & 1;
        if (k0 + KB3 < GN) fill(buf ^ 1, k0 + KB3);   // overlap next fill with compute
        __builtin_prefetch(arow + k0 + KB3, 0, 1);
        #pragma unroll 4
        for (int kk = 0; kk < KB3; kk += 4) {
            v2f a = *(const v2f*)(arow + k0 + kk);
            int kb = kk + kh;
            #pragma unroll
            for (int t = 0; t < 4; ++t) {
                v2f bf;
                bf.x = lz[buf][kb][t * 16 + n];
                bf.y = lz[buf][kb + 1][t * 16 + n];
                acc[t] = __builtin_amdgcn_wmma_f32_16x16x4_f32(
                    false, a, false, bf, (short)0, acc[t], false, false);
            }
        }
        wait_async0();
        __syncthreads();
    }
    #pragma unroll
    for (int t = 0; t < 4; ++t)
        #pragma unroll
        for (int r = 0; r < 8; ++r) {
            int row = row0 + r + 8 * (lane >> 4);
            int col = t * 16 + n;
            size_t gi = ((size_t)b * GN + row) * GD + col;
            float self = zbase[(size_t)row * GD + col];
            float y = dis[b * GN + row] * (acc[t][r] + self) + b1[col];
            y = y > 0.0f ? y : 0.0f;
            unsigned h = hashu((unsigned)gi * 2654435761u + 0x9E3779B9u);
            Hd[gi] = (h & 0x10000u) ? y * 2.0f : 0.0f;   // p=0.5, scale 1/(1-p)
        }
}

// ---------------------------------------------------------------------------
// K4: Z2p = rowscale(dis) * (Hd @ W2)   M=16384 K=64 N=16
// ---------------------------------------------------------------------------
__global__ __launch_bounds__(256) void k_hw2(const float* __restrict__ Hd,
                                             const float* __restrict__ W2,
                                             const float* __restrict__ dis,
                                             float* __restrict__ Z2p) {
    const int lane = threadIdx.x & 31;
    const int wv   = threadIdx.x >> 5;
    const int row0 = (blockIdx.x * 8 + wv) * 16;
    const int m  = lane & 15;
    const int kh = (lane >> 4) * 2;
    const int n  = lane & 15;
    const float* hrow = Hd + (size_t)(row0 + m) * GD + kh;
    v8f acc = {};
    for (int k = 0; k < GD; k += 4) {
        v2f a = *(const v2f*)(hrow + k);
        v2f bf;
        bf.x = W2[(k + kh)     * GC + n];
        bf.y = W2[(k + kh + 1) * GC + n];
        acc = __builtin_amdgcn_wmma_f32_16x16x4_f32(
            false, a, false, bf, (short)0, acc, false, false);
    }
    #pragma unroll
    for (int r = 0; r < 8; ++r) {
        int row = row0 + r + 8 * (lane >> 4);
        Z2p[(size_t)row * GC + n] = acc[r] * dis[row];
    }
}

// ---------------------------------------------------------------------------
// K5: out = dis*(A@Z2p + Z2p) + b2
//     per batch: 2048x2048 @ 2048x16, double-buffered async LDS slabs
// ---------------------------------------------------------------------------
#define KB5 256
__global__ __launch_bounds__(256) void k_layer2(const float* __restrict__ A,
                                                const float* __restrict__ Z2p,
                                                const float* __restrict__ dis,
                                                const float* __restrict__ b2,
                                                float* __restrict__ out) {
    __shared__ float lz[2][KB5 * GC];             // stride 16: half-waves 32 banks apart
    const int b     = blockIdx.x >> 4;
    const int mtile = blockIdx.x & 15;
    const int lane  = threadIdx.x & 31;
    const int wv    = threadIdx.x >> 5;
    const int row0  = mtile * 128 + wv * 16;
    const int m  = lane & 15;
    const int kh = (lane >> 4) * 2;
    const int n  = lane & 15;
    const float* arow  = A   + ((size_t)b * GN + row0 + m) * GN + kh;
    const float* zbase = Z2p + (size_t)b * GN * GC;
    const unsigned lbase = (unsigned)(uintptr_t)(void*)&lz[0][0];

    auto fill = [&](int buf, int k0) {
        const float* src = zbase + (size_t)k0 * GC;
        const unsigned lb = lbase + (unsigned)buf * (KB5 * GC * 4);
        #pragma unroll
        for (int it = 0; it < 4; ++it) {          // 1024 x b128 per slab, 4 per thread
            int idx = (int)threadIdx.x + it * 256;
            async_g2l_b128(lb + (unsigned)(idx * 16), src + idx * 4);
        }
    };

    v8f acc = {};
    fill(0, 0);
    wait_async0();
    __syncthreads();
    for (int k0 = 0; k0 < GN; k0 += KB5) {
        const int buf = (k0 / KB5) & 1;
        if (k0 + KB5 < GN) fill(buf ^ 1, k0 + KB5);
        __builtin_prefetch(arow + k0 + KB5, 0, 1);
        #pragma unroll 4
        for (int kk = 0; kk < KB5; kk += 4) {
            v2f a = *(const v2f*)(arow + k0 + kk);
            int kb = kk + kh;
            v2f bf;
            bf.x = lz[buf][kb * GC + n];
            bf.y = lz[buf][(kb + 1) * GC + n];
            acc = __builtin_amdgcn_wmma_f32_16x16x4_f32(
                false, a, false, bf, (short)0, acc, false, false);
        }
        wait_async0();
        __syncthreads();
    }
    #pragma unroll
    for (int r = 0; r < 8; ++r) {
        int row = row0 + r + 8 * (lane >> 4);
        float self = zbase[(size_t)row * GC + n];
        out[((size_t)b * GN + row) * GC + n] =
            dis[b * GN + row] * (acc[r] + self) + b2[n];
    }
}

// ---------------------------------------------------------------------------
extern "C" void kernel_launch(void* const* d_in, const int* in_sizes, int n_in,
                              void* d_out, int out_size, void* d_ws, size_t ws_size,
                              hipStream_t stream) {
    const float* X  = (const float*)d_in[0];
    const float* A  = (const float*)d_in[1];
    const float* W1 = (const float*)d_in[2];
    const float* b1 = (const float*)d_in[3];
    const float* W2 = (const float*)d_in[4];
    const float* b2 = (const float*)d_in[5];
    float* out = (float*)d_out;

    float* dis = (float*)d_ws;            // B*N           = 16384 f32
    float* Z1p = dis + GB * GN;           // B*N*H         = 1048576 f32
    float* Hd  = Z1p + GB * GN * GD;      // B*N*H         = 1048576 f32
    float* Z2p = Hd  + GB * GN * GD;      // B*N*C         = 262144 f32

    k_degree<<<GB * GN / 8, 256, 0, stream>>>(A, dis);
    k_xw1   <<<GB * GN / 128, 256, 0, stream>>>(X, W1, dis, Z1p);
    k_layer1<<<GB * (GN / 128), 256, 0, stream>>>(A, Z1p, dis, b1, Hd);
    k_hw2   <<<GB * GN / 128, 256, 0, stream>>>(Hd, W2, dis, Z2p);
    k_layer2<<<GB * (GN / 128), 256, 0, stream>>>(A, Z2p, dis, b2, out);
}